// CustomAttention_61649960567300
// MI455X (gfx1250) — compile-verified
//
#include <hip/hip_runtime.h>

#define NDIM 4096
#define NBATCH 16

typedef __attribute__((ext_vector_type(2))) float v2f;
typedef __attribute__((ext_vector_type(8))) float v8f;

__device__ __forceinline__ float fast_tanh(float v) {
#if __has_builtin(__builtin_amdgcn_tanhf)
  return __builtin_amdgcn_tanhf(v);   // v_tanh_f32 (gfx1250 TRANS op)
#else
  return tanhf(v);
#endif
}

// One wave32 computes a 16-row output tile out[batch, tile*16 .. tile*16+15].
// For each K=4 slab of m:
//   A (16x4 f32): A[n][k] = tanh(a * x[b,n] * x[b,m+k])
//   B (4x16 f32): B[k][j] = x[b,m+k]  (same value in every column j)
//   acc = WMMA(A,B,acc)  -> every column of acc holds sum_m tanh(..)*x[b,m]
__global__ __launch_bounds__(256) void CustomAttention_tanh_wmma_kernel(
    const float* __restrict__ x,     // [16, 4096]
    const float* __restrict__ a,     // [1]
    const float* __restrict__ bsc,   // [1]
    float* __restrict__ out)         // [16, 4096]
{
  __shared__ float xrow[NDIM];       // 16 KB: one batch row, shared by 8 waves

  const int lane  = threadIdx.x & 31;
  const int wave  = threadIdx.x >> 5;
  const int gw    = blockIdx.x * 8 + wave;   // 0 .. 4095
  const int batch = gw >> 8;                 // 256 tiles per batch; 8 | 256 so
  const int tile  = gw & 255;                // all 8 waves in a block share `batch`

  // Cooperative stage of x[batch, :] into LDS (uniform trip count: 4 iters)
  const float* __restrict__ xb = x + batch * NDIM;
  for (int i = threadIdx.x * 4; i < NDIM; i += 256 * 4) {
    *(float4*)(xrow + i) = *(const float4*)(xb + i);
  }
  __syncthreads();

  const int half = lane >> 4;        // 0: K=0,1   1: K=2,3
  const int row  = lane & 15;        // M index for A
  const float axn = a[0] * xrow[tile * 16 + row];

  v8f acc = {};
  #pragma unroll 8
  for (int m = 0; m < NDIM; m += 4) {
    const float2 xm = *(const float2*)(xrow + m + 2 * half);  // ds_load_b64
    v2f Bv; Bv.x = xm.x;                Bv.y = xm.y;
    v2f Av; Av.x = fast_tanh(axn * xm.x); Av.y = fast_tanh(axn * xm.y);
    // 8 args: (neg_a, A, neg_b, B, c_mod, C, reuse_a, reuse_b)
    acc = __builtin_amdgcn_wmma_f32_16x16x4_f32(
        false, Av, false, Bv, (short)0, acc, false, false);
  }

  // D layout: VGPR r, lane L -> M = r + 8*(L/16), N = L%16. All columns equal,
  // so lanes with N==0 (lanes 0 and 16) hold the 16 results in acc[0..7].
  if (row == 0) {
    const float bs = bsc[0];
    float* __restrict__ o = out + batch * NDIM + tile * 16 + 8 * half;
    #pragma unroll
    for (int r = 0; r < 8; ++r) o[r] = bs * acc[r];
  }
}

extern "C" void kernel_launch(void* const* d_in, const int* in_sizes, int n_in,
                              void* d_out, int out_size, void* d_ws, size_t ws_size,
                              hipStream_t stream) {
  const float* x   = (const float*)d_in[0];
  const float* a   = (const float*)d_in[1];
  const float* bsc = (const float*)d_in[2];
  float* out       = (float*)d_out;

  // 4096 waves total: 512 blocks x 8 waves (256 threads)
  dim3 grid(512), block(256);
  CustomAttention_tanh_wmma_kernel<<<grid, block, 0, stream>>>(x, a, bsc, out);
}